// SkeletonConvolution_90031104459329
// MI455X (gfx1250) — compile-verified
//
#include <hip/hip_runtime.h>

// ---------------------------------------------------------------------------
// SkeletonConvolution for MI455X (gfx1250, wave32, WMMA)
//
// out[b,co,k,t] = bias[co] + sum_{ci=0..15, dt=0..2} W[co,ci,dt] * x[b,ci,idx[k],t+dt-1]
// B=32, C_IN=C_OUT=16, J=24, K=75 (gathered joints), T=4096, fp32.
//
// Memory-bound (~830MB traffic, ~36us floor @ 23.3TB/s). Cast as GEMM:
//   M=16 (co), Kdim=48 (ci*3+dt im2col), N = 32*75*4096,
// computed with chained V_WMMA_F32_16X16X4_F32 (fp32 end-to-end).
// ---------------------------------------------------------------------------

typedef __attribute__((ext_vector_type(2))) float v2f;
typedef __attribute__((ext_vector_type(8))) float v8f;

namespace {
constexpr int kB    = 32;
constexpr int kCin  = 16;
constexpr int kCout = 16;
constexpr int kJ    = 24;
constexpr int kT    = 4096;
constexpr int kK    = 75;    // total gathered joint rows (len(ALL_IDX))
constexpr int kTileT   = 128;   // t-columns per workgroup (8 waves x 16)
constexpr int kThreads = 256;   // 8 wave32
constexpr int kLdsCols = 132;   // 130 used (+2 pad vs bank conflicts)
}

// ALL_IDX from the reference's _neighbor_lists() (computed offline, fixed).
__constant__ int c_all_idx[kK] = {
    0,1,2,3,10,11,15,22,23,          // nl[0] (root + end effectors)
    0,1,4,  0,2,5,  0,3,6,
    1,4,7,  2,5,8,  3,6,9,
    4,7,10, 5,8,11,
    6,9,12,13,14,                    // nl[9]
    7,10,   8,11,
    9,12,15, 9,13,16, 9,14,17,
    12,15,
    13,16,18, 14,17,19, 16,18,20,
    17,19,21, 18,20,22, 19,21,23,
    20,22,  21,23
};

// LDS flat offset for im2col row kk = ci*3+dt at tap column:
__host__ __device__ constexpr int lds_off(int kk) {
    return (kk / 3) * kLdsCols + (kk % 3);
}

__global__ void __launch_bounds__(kThreads)
skel_conv_wmma(const float* __restrict__ x,     // [32,16,24,4096]
               const float* __restrict__ Wg,    // [16,16,1,3] -> [co][ci*3+dt]
               const float* __restrict__ bias,  // [16]
               float* __restrict__ out)         // [32,16,75,4096]
{
    __shared__ float xs[kCin * kLdsCols];

    const int tchunk = blockIdx.x;          // 0..31  (T / 128)
    const int kj     = blockIdx.y;          // 0..74  gathered joint row
    const int bb     = blockIdx.z;          // 0..31  batch (slowest -> L2 reuse)
    const int j      = c_all_idx[kj];

    const int tid  = threadIdx.x;
    const int lane = tid & 31;
    const int wave = tid >> 5;
    const int half = lane >> 4;             // 0: lanes 0-15, 1: lanes 16-31
    const int ln   = lane & 15;             // M (co) for A, N (t) for B/C/D

    const int tbase = tchunk * kTileT;

    // ---- Stage input slab into LDS: x[bb, ci, j, tbase-1 .. tbase+128] ----
    // 16 rows x 130 cols, zero-padded at the T boundary (conv pad=1).
    const float* xrow = x + (((size_t)bb * kCin) * kJ + j) * (size_t)kT;
    for (int i = tid; i < kCin * 130; i += kThreads) {
        const int ci = i / 130;
        const int tp = i - ci * 130;
        const int t  = tbase - 1 + tp;
        float v = 0.0f;
        if (t >= 0 && t < kT)
            v = xrow[(size_t)ci * kJ * kT + t];
        xs[ci * kLdsCols + tp] = v;
    }

    // ---- A fragments: weights, 16x48, 12 steps of K=4 ----
    // 32-bit A 16x4 layout: lanes 0-15 -> K = 4s,4s+1 ; lanes 16-31 -> K = 4s+2,4s+3
    v2f a[12];
#pragma unroll
    for (int s = 0; s < 12; ++s) {
        const int kk = 4 * s + 2 * half;
        a[s].x = Wg[ln * 48 + kk];
        a[s].y = Wg[ln * 48 + kk + 1];
    }

    __syncthreads();

    // ---- 12 chained WMMA accumulations: D = A(16x4) * B(4x16) + C ----
    const int col = wave * 16 + ln;         // local t-column (0..127)
    v8f c = {};
#pragma unroll
    for (int s = 0; s < 12; ++s) {
        // B 4x16 layout mirrors A: lanes 0-15 -> K rows 4s,4s+1 ; 16-31 -> 4s+2,4s+3
        const int o0 = half ? lds_off(4 * s + 2) : lds_off(4 * s);
        const int o1 = half ? lds_off(4 * s + 3) : lds_off(4 * s + 1);
        v2f bfrag;
        bfrag.x = xs[o0 + col];
        bfrag.y = xs[o1 + col];
        c = __builtin_amdgcn_wmma_f32_16x16x4_f32(
                /*neg_a=*/false, a[s], /*neg_b=*/false, bfrag,
                /*c_mod=*/(short)0, c, /*reuse_a=*/false, /*reuse_b=*/false);
    }

    // ---- Bias + store ----
    // C/D 16x16 f32 layout: VGPR r -> co = r (lanes 0-15) / r+8 (lanes 16-31); N = t.
    const int t_out = tbase + col;
#pragma unroll
    for (int r = 0; r < 8; ++r) {
        const int co = r + 8 * half;
        const float v = c[r] + bias[co];
        out[(((size_t)bb * kCout + co) * kK + kj) * (size_t)kT + t_out] = v;
    }
}

extern "C" void kernel_launch(void* const* d_in, const int* in_sizes, int n_in,
                              void* d_out, int out_size, void* d_ws, size_t ws_size,
                              hipStream_t stream) {
    const float* x  = (const float*)d_in[0];   // 32*16*24*4096
    const float* Wg = (const float*)d_in[1];   // 16*16*1*3
    const float* b  = (const float*)d_in[2];   // 16
    float* out = (float*)d_out;                // 32*16*75*4096

    dim3 grid(kT / kTileT, kK, kB);            // (32, 75, 32); b slowest for L2 reuse
    skel_conv_wmma<<<grid, kThreads, 0, stream>>>(x, Wg, b, out);
}